// WriteMemory_27762668601944
// MI455X (gfx1250) — compile-verified
//
#include <hip/hip_runtime.h>
#include <hip/hip_bf16.h>

typedef __attribute__((ext_vector_type(2))) float v2f;
typedef __attribute__((ext_vector_type(8))) float v8f;

// Problem constants (from reference setup_inputs)
#define BB 32
#define DD 1024
#define DA 64
#define DM 128
#define HH 8
#define NN 256
#define WCOLS 208   // 128 (mat_v) + 64 (mat_k) + 1 (norm) + 15 pad -> 13 WMMA n-tiles
#define EPS 1e-5f
#define MAXWF 0.7f

// ---------------------------------------------------------------------------
// K1: projections  kp = key@Wk.T + bk  (32x512),  vp = values@Wv.T + bv (32x1024)
// One wave per 16x16 output tile, K=1024 in steps of 4 (V_WMMA_F32_16X16X4_F32).
// blockDim=64 (2 waves = 2 m-tiles), gridDim=96 (32 kp n-tiles + 64 vp n-tiles)
// ---------------------------------------------------------------------------
__global__ void __launch_bounds__(64) proj_kernel(
    const float* __restrict__ key, const float* __restrict__ values,
    const float* __restrict__ Wk, const float* __restrict__ bk,
    const float* __restrict__ Wv, const float* __restrict__ bv,
    float* __restrict__ kp, float* __restrict__ vp) {
  const int lane = threadIdx.x & 31;
  const int m0 = (threadIdx.x >> 5) * 16;      // m-tile per wave (rows = batch)
  const int nt = blockIdx.x;
  const float* src; const float* W; const float* bias; float* dst; int n0, ldD;
  if (nt < 32) { src = key;    W = Wk; bias = bk; dst = kp; n0 = nt * 16;        ldD = HH * DA; }
  else         { src = values; W = Wv; bias = bv; dst = vp; n0 = (nt - 32) * 16; ldD = HH * DM; }
  const int lm = lane & 15;
  const int kh = lane >> 4;
  // A: src[m0+lm][kb + 2*kh + g] ; B(=W.T): W[n0+lm][kb + 2*kh + g]  -> float2 loads
  const float2* srow = (const float2*)(src + (size_t)(m0 + lm) * DD);
  const float2* wrow = (const float2*)(W + (size_t)(n0 + lm) * DD);
  v8f c = {};
  for (int kb = 0; kb < DD; kb += 4) {
    float2 af = srow[(kb >> 1) + kh];
    float2 bf = wrow[(kb >> 1) + kh];
    v2f a = {af.x, af.y};
    v2f b = {bf.x, bf.y};
    c = __builtin_amdgcn_wmma_f32_16x16x4_f32(false, a, false, b, (short)0, c,
                                              false, false);
  }
  const float bb = bias[n0 + lm];
#pragma unroll
  for (int g = 0; g < 8; ++g) {
    int m = m0 + g + 8 * kh;                   // C/D layout: m = g + 8*(lane/16)
    dst[(size_t)m * ldD + n0 + lm] = c[g] + bb;
  }
}

// ---------------------------------------------------------------------------
// K2: per-batch Wext (64 x 208): [relu(k)^T v | relu(k)^T k | relu(k).sum | 0]
// K-reduction over H=8 heads; k/v rows staged in LDS. gridDim=B, blockDim=256.
// ---------------------------------------------------------------------------
__global__ void __launch_bounds__(256) build_wext(
    const float* __restrict__ kp, const float* __restrict__ vp,
    float* __restrict__ Wext) {
  const int b = blockIdx.x;
  __shared__ float skp[HH * DA];   // raw k (v-operand of second attention)
  __shared__ float ska[HH * DA];   // relu(k)
  __shared__ float svp[HH * DM];
  for (int i = threadIdx.x; i < HH * DA; i += 256) {
    float x = kp[(size_t)b * HH * DA + i];
    skp[i] = x;
    ska[i] = fmaxf(x, 0.f);
  }
  for (int i = threadIdx.x; i < HH * DM; i += 256)
    svp[i] = vp[(size_t)b * HH * DM + i];
  __syncthreads();
  for (int idx = threadIdx.x; idx < DA * WCOLS; idx += 256) {
    const int a = idx / WCOLS, col = idx % WCOLS;
    float acc = 0.f;
    if (col < DM) {
#pragma unroll
      for (int h = 0; h < HH; ++h) acc += ska[h * DA + a] * svp[h * DM + col];
    } else if (col < DM + DA) {
      const int j = col - DM;
#pragma unroll
      for (int h = 0; h < HH; ++h) acc += ska[h * DA + a] * skp[h * DA + j];
    } else if (col == DM + DA) {
#pragma unroll
      for (int h = 0; h < HH; ++h) acc += ska[h * DA + a];
    }
    Wext[((size_t)b * DA + a) * WCOLS + col] = acc;
  }
}

// ---------------------------------------------------------------------------
// K3: per-batch GEMM  numo[b] (256 x 208) = relu(addresses)(256x64) @ Wext[b]
// Shared numerators for both linear attentions + the shared denominator col.
// grid = B*16 (one block per (b, m-tile)), blockDim=128 (4 waves over 13 n-tiles)
// ---------------------------------------------------------------------------
__global__ void __launch_bounds__(128) attn_gemm(
    const float* __restrict__ addresses, const float* __restrict__ Wext,
    float* __restrict__ numo) {
  const int blk = blockIdx.x;
  const int b = blk >> 4;
  const int m0 = (blk & 15) * 16;
  const int lane = threadIdx.x & 31;
  const int wave = threadIdx.x >> 5;
  const int lm = lane & 15;
  const int kh = lane >> 4;
  const float2* arow = (const float2*)(addresses + (size_t)(m0 + lm) * DA);
  const float* Wb = Wext + (size_t)b * DA * WCOLS;
  for (int ntile = wave; ntile < WCOLS / 16; ntile += 4) {
    const int n0 = ntile * 16;
    v8f c = {};
    for (int kb = 0; kb < DA; kb += 4) {
      float2 af = arow[(kb >> 1) + kh];
      v2f a = {fmaxf(af.x, 0.f), fmaxf(af.y, 0.f)};      // qa = relu(addresses)
      const int k0 = kb + 2 * kh;
      v2f bf = {Wb[(size_t)k0 * WCOLS + n0 + lm],
                Wb[(size_t)(k0 + 1) * WCOLS + n0 + lm]};
      c = __builtin_amdgcn_wmma_f32_16x16x4_f32(false, a, false, bf, (short)0, c,
                                                false, false);
    }
#pragma unroll
    for (int g = 0; g < 8; ++g) {
      int m = m0 + g + 8 * kh;
      numo[((size_t)b * NN + m) * WCOLS + n0 + lm] = c[g];
    }
  }
}

// ---------------------------------------------------------------------------
// K4: wave-per-(b,n) epilogue: normalize by shared den, write logit via lane
// reduction, sigmoids, emit wf / relu(key_mem) / value_mem + normalizer & mass.
// batch_mask folds into wf=0 (exact masked semantics). grid=1024, blockDim=256.
// ---------------------------------------------------------------------------
__global__ void __launch_bounds__(256) epilogue_kernel(
    const float* __restrict__ numo, const float* __restrict__ interp_logits,
    const float* __restrict__ Ww, const float* __restrict__ bw,
    const float* __restrict__ normalizer, const float* __restrict__ write_mass,
    const unsigned char* __restrict__ batch_mask,
    float* __restrict__ value_mem, float* __restrict__ wf,
    float* __restrict__ kam, float* __restrict__ out_norm,
    float* __restrict__ out_mass) {
  const int wid = blockIdx.x * 8 + (threadIdx.x >> 5);  // 0..B*N-1
  const int lane = threadIdx.x & 31;
  const int b = wid >> 8;
  const int n = wid & (NN - 1);
  const float* row = numo + (size_t)wid * WCOLS;
  const float inv = 1.f / (row[DM + DA] + EPS);          // shared denominator
#pragma unroll
  for (int j = 0; j < 4; ++j) {                          // value_mem (128 cols)
    int m = lane + j * 32;
    value_mem[(size_t)wid * DM + m] = row[m] * inv;
  }
  float km0 = row[DM + lane] * inv;                      // key_mem (64 cols)
  float km1 = row[DM + 32 + lane] * inv;
  float p = km0 * Ww[lane] + km1 * Ww[lane + 32];        // key_mem @ Ww.T
  for (int off = 16; off >= 1; off >>= 1) p += __shfl_xor(p, off, 32);
  const float wp = 1.f / (1.f + __expf(-(p + bw[0])));
  const float wps = (batch_mask[b] != 0) ? 0.f : wp;     // masked -> no write
  const float* il = interp_logits + (size_t)n * DA;
  const float kms[2] = {km0, km1};
#pragma unroll
  for (int j = 0; j < 2; ++j) {
    const int a = lane + j * 32;
    const float ip = 1.f / (1.f + __expf(-il[a]));
    const float w = MAXWF * wps * ip;
    const float ka = fmaxf(kms[j], 0.f);
    const size_t idx = (size_t)wid * DA + a;
    wf[idx] = w;
    kam[idx] = ka;
    out_norm[idx] = normalizer[idx] * (1.f - w) + ka * w;
  }
  if (lane == 0) out_mass[wid] = write_mass[wid] + wps;
}

// ---------------------------------------------------------------------------
// K5: THE bandwidth kernel — single pass over matrix (256MB in + 256MB out).
// One workgroup per (b,n); LDS stages (1-wf), ka*wf, value_mem; b128 traffic.
// grid = B*N = 8192, blockDim = 256 (8 waves / WGP-friendly).
// ---------------------------------------------------------------------------
__global__ void __launch_bounds__(256) update_matrix_kernel(
    const float* __restrict__ matrix, const float* __restrict__ value_mem,
    const float* __restrict__ wf, const float* __restrict__ kam,
    float* __restrict__ out_matrix) {
  const int bn = blockIdx.x;
  __shared__ float s_om[DA];     // 1 - wf
  __shared__ float s_kw[DA];     // ka * wf
  __shared__ float4 s_vm[DM / 4];
  const int t = threadIdx.x;
  if (t < DA) {
    float w = wf[(size_t)bn * DA + t];
    s_om[t] = 1.f - w;
    s_kw[t] = kam[(size_t)bn * DA + t] * w;
  } else if (t < DA + DM / 4) {
    s_vm[t - DA] = ((const float4*)(value_mem + (size_t)bn * DM))[t - DA];
  }
  __syncthreads();
  const float4* min4 = (const float4*)matrix + (size_t)bn * (DA * DM / 4);
  float4* mout4 = (float4*)out_matrix + (size_t)bn * (DA * DM / 4);
#pragma unroll
  for (int j = 0; j < (DA * DM / 4) / 256; ++j) {        // 8 iterations
    const int idx4 = t + j * 256;
    const int a = idx4 >> 5;                             // dA row
    const int m4 = idx4 & 31;                            // dM float4 col
    const float4 mv = min4[idx4];
    const float om = s_om[a], kw = s_kw[a];
    const float4 vm = s_vm[m4];
    float4 o;
    o.x = mv.x * om + vm.x * kw;
    o.y = mv.y * om + vm.y * kw;
    o.z = mv.z * om + vm.z * kw;
    o.w = mv.w * om + vm.w * kw;
    mout4[idx4] = o;
  }
}

extern "C" void kernel_launch(void* const* d_in, const int* in_sizes, int n_in,
                              void* d_out, int out_size, void* d_ws, size_t ws_size,
                              hipStream_t stream) {
  const float* key           = (const float*)d_in[0];
  const float* values        = (const float*)d_in[1];
  const float* addresses     = (const float*)d_in[2];
  const float* matrix        = (const float*)d_in[3];
  const float* normalizer    = (const float*)d_in[4];
  const float* write_mass    = (const float*)d_in[5];
  const float* Wk            = (const float*)d_in[6];
  const float* bk            = (const float*)d_in[7];
  const float* Wv            = (const float*)d_in[8];
  const float* bv            = (const float*)d_in[9];
  const float* interp_logits = (const float*)d_in[10];
  const float* Ww            = (const float*)d_in[11];
  const float* bw            = (const float*)d_in[12];
  const unsigned char* batch_mask = (const unsigned char*)d_in[13];

  // workspace carve-up (floats): ~17.1 MB total
  float* ws        = (float*)d_ws;
  float* kp        = ws;                         // B*H*dA      = 16384
  float* vp        = kp + BB * HH * DA;          // B*H*dM      = 32768
  float* Wext      = vp + BB * HH * DM;          // B*dA*208    = 425984
  float* numo      = Wext + BB * DA * WCOLS;     // B*N*208     = 1703936
  float* value_mem = numo + (size_t)BB * NN * WCOLS;   // B*N*dM = 1048576
  float* wfb       = value_mem + (size_t)BB * NN * DM; // B*N*dA = 524288
  float* kam       = wfb + (size_t)BB * NN * DA;       // B*N*dA = 524288

  // outputs: new_matrix | new_normalizer | new_mass (flat, in return order)
  float* out_matrix = (float*)d_out;
  float* out_norm   = out_matrix + (size_t)BB * NN * DA * DM;
  float* out_mass   = out_norm + (size_t)BB * NN * DA;

  proj_kernel<<<96, 64, 0, stream>>>(key, values, Wk, bk, Wv, bv, kp, vp);
  build_wext<<<BB, 256, 0, stream>>>(kp, vp, Wext);
  attn_gemm<<<BB * (NN / 16), 128, 0, stream>>>(addresses, Wext, numo);
  epilogue_kernel<<<BB * NN / 8, 256, 0, stream>>>(
      numo, interp_logits, Ww, bw, normalizer, write_mass, batch_mask,
      value_mem, wfb, kam, out_norm, out_mass);
  update_matrix_kernel<<<BB * NN, 256, 0, stream>>>(matrix, value_mem, wfb, kam,
                                                    out_matrix);
}